// GraphConvolution_12790412607564
// MI455X (gfx1250) — compile-verified
//
#include <hip/hip_runtime.h>

typedef __attribute__((ext_vector_type(2))) float v2f;
typedef __attribute__((ext_vector_type(8))) float v8f;

#define DD 128   // input feature dim D
#define HH 128   // hidden dim H

// ---------------------------------------------------------------------------
// Wh[M x H] = feat[M x D] @ W[D x H] + b[H]
// One 256-thread block (8 waves) per 16-row strip. A tile staged in LDS
// (row-padded to 132 floats -> conflict-free reads). Each wave computes one
// 16x16 output tile via 32 steps of V_WMMA_F32_16X16X4_F32 (exact fp32 path).
// Epilogue: single wave-uniform scalar branch; full tiles store branch-free.
// ---------------------------------------------------------------------------
__global__ __launch_bounds__(256) void gemm_bias_wmma(
    const float* __restrict__ feat, const float* __restrict__ W,
    const float* __restrict__ bias, float* __restrict__ Wh, int M)
{
    __shared__ float As[16][132];

    const int m0  = blockIdx.x * 16;
    const int tid = threadIdx.x;

    // Cooperative, coalesced load of the 16x128 A tile (clamp rows for tails).
    for (int i = tid; i < 16 * DD; i += 256) {
        int r = i >> 7;          // i / 128
        int c = i & 127;         // i % 128
        int rg = m0 + r;
        if (rg >= M) rg = M - 1; // harmless clamp; tail rows never stored
        As[r][c] = feat[(size_t)rg * DD + c];
    }
    __syncthreads();

    const int wave = tid >> 5;
    const int lane = tid & 31;
    const int half = lane >> 4;      // 0: lanes 0-15, 1: lanes 16-31
    const int mrow = lane & 15;      // A row within tile
    const int col  = lane & 15;      // B/C/D column within tile
    const int n0   = wave * 16;      // this wave's N tile

    // C/D layout: every VGPR j holds N = lane&15; bias depends only on column.
    v8f acc;
    const float bv = bias[n0 + col];
    #pragma unroll
    for (int j = 0; j < 8; ++j) acc[j] = bv;

    // K loop: 4 K-values per WMMA.
    // A 16x4 f32 layout: VGPR0 = {K=k0 (lanes 0-15), K=k0+2 (lanes 16-31)},
    //                    VGPR1 = {K=k0+1,            K=k0+3}.
    // B 4x16 mirrors the K split with N across lanes 0-15.
    #pragma unroll 4
    for (int k0 = 0; k0 < DD; k0 += 4) {
        const int ka = k0 + 2 * half;
        v2f a, bfr;
        a[0]   = As[mrow][ka + 0];
        a[1]   = As[mrow][ka + 1];
        bfr[0] = W[(size_t)(ka + 0) * HH + n0 + col];
        bfr[1] = W[(size_t)(ka + 1) * HH + n0 + col];
        acc = __builtin_amdgcn_wmma_f32_16x16x4_f32(
            /*neg_a=*/false, a, /*neg_b=*/false, bfr,
            /*c_mod=*/(short)0, acc, /*reuse_a=*/false, /*reuse_b=*/false);
    }

    // D layout: VGPR j -> M = j (lanes 0-15) / M = 8+j (lanes 16-31), N = lane&15.
    // m0 and M are wave-uniform -> this is one scalar branch, not exec juggling.
    float* __restrict__ outp = Wh + (size_t)(m0 + 8 * half) * HH + n0 + col;
    if (m0 + 16 <= M) {
        #pragma unroll
        for (int j = 0; j < 8; ++j)
            outp[(size_t)j * HH] = acc[j];
    } else {
        #pragma unroll
        for (int j = 0; j < 8; ++j)
            if (m0 + 8 * half + j < M)
                outp[(size_t)j * HH] = acc[j];
    }
}

// ---------------------------------------------------------------------------
// One wave per edge: gather a 128-float row (float4 per lane, coalesced),
// scatter with non-returning f32 atomics; lane 0 counts the in-degree.
// Wh (102 MB for both relations) is L2-resident (192 MB), so random row
// gathers and the atomic adds mostly stay on-chip.
// ---------------------------------------------------------------------------
__global__ __launch_bounds__(256) void scatter_mean_accum(
    const float* __restrict__ Wh_src, const int* __restrict__ src,
    const int* __restrict__ dst, float* __restrict__ out_sum,
    float* __restrict__ cnt, int E)
{
    const int eid  = (int)((blockIdx.x * (unsigned)blockDim.x + threadIdx.x) >> 5);
    const int lane = threadIdx.x & 31;
    if (eid >= E) return;

    const int s = src[eid];
    const int d = dst[eid];

    const float4 v = ((const float4*)(Wh_src + (size_t)s * HH))[lane];
    float* o = out_sum + (size_t)d * HH + lane * 4;
    atomicAdd(o + 0, v.x);
    atomicAdd(o + 1, v.y);
    atomicAdd(o + 2, v.z);
    atomicAdd(o + 3, v.w);
    if (lane == 0) atomicAdd(cnt + d, 1.0f);
}

// ---------------------------------------------------------------------------
// out[n,h] = sum[n,h] / max(cnt[n], 1)   (zero in-degree -> 0, DGL semantics)
// float4 per thread; exact IEEE divides to match the reference numerics.
// ---------------------------------------------------------------------------
__global__ __launch_bounds__(256) void normalize_mean(
    float* __restrict__ out, const float* __restrict__ cnt, size_t total4)
{
    const size_t i = (size_t)blockIdx.x * blockDim.x + threadIdx.x;
    if (i >= total4) return;
    float4 v = ((float4*)out)[i];
    const float c = fmaxf(cnt[(i * 4) >> 7], 1.0f); // H == 128
    v.x /= c; v.y /= c; v.z /= c; v.w /= c;
    ((float4*)out)[i] = v;
}

// ---------------------------------------------------------------------------
extern "C" void kernel_launch(void* const* d_in, const int* in_sizes, int n_in,
                              void* d_out, int out_size, void* d_ws, size_t ws_size,
                              hipStream_t stream) {
    const float* feat_A = (const float*)d_in[0];
    const float* feat_B = (const float*)d_in[1];
    const float* W_ab   = (const float*)d_in[2];
    const float* b_ab   = (const float*)d_in[3];
    const float* W_ba   = (const float*)d_in[4];
    const float* b_ba   = (const float*)d_in[5];
    const int*   src_ab = (const int*)d_in[6];
    const int*   dst_ab = (const int*)d_in[7];
    const int*   src_ba = (const int*)d_in[8];
    const int*   dst_ba = (const int*)d_in[9];

    const int NA = in_sizes[0] / DD;   // 100000
    const int NB = in_sizes[1] / DD;   // 100000
    const int E  = in_sizes[6];        // 600000

    // Workspace: Wh_A | Wh_B | cnt_B | cnt_A  (~103 MB)
    float* Wh_A  = (float*)d_ws;
    float* Wh_B  = Wh_A + (size_t)NA * HH;
    float* cnt_B = Wh_B + (size_t)NB * HH;
    float* cnt_A = cnt_B + NB;

    float* h_A = (float*)d_out;                // [NA, H]
    float* h_B = h_A + (size_t)NA * HH;        // [NB, H]

    // Zero accumulators (graph-capture-safe async memsets).
    hipMemsetAsync(d_out, 0, (size_t)out_size * sizeof(float), stream);
    hipMemsetAsync(cnt_B, 0, ((size_t)NA + NB) * sizeof(float), stream);

    // Per-relation linear projections (WMMA fp32).
    gemm_bias_wmma<<<(NA + 15) / 16, 256, 0, stream>>>(feat_A, W_ab, b_ab, Wh_A, NA);
    gemm_bias_wmma<<<(NB + 15) / 16, 256, 0, stream>>>(feat_B, W_ba, b_ba, Wh_B, NB);

    // Edge-parallel scatter (one wave32 per edge).
    const int sblocks = (E * 32 + 255) / 256;
    scatter_mean_accum<<<sblocks, 256, 0, stream>>>(Wh_A, src_ab, dst_ab, h_B, cnt_B, E);
    scatter_mean_accum<<<sblocks, 256, 0, stream>>>(Wh_B, src_ba, dst_ba, h_A, cnt_A, E);

    // Mean normalization (float4 granularity).
    const size_t tot4A = (size_t)NA * HH / 4, tot4B = (size_t)NB * HH / 4;
    normalize_mean<<<(unsigned)((tot4A + 255) / 256), 256, 0, stream>>>(h_A, cnt_A, tot4A);
    normalize_mean<<<(unsigned)((tot4B + 255) / 256), 256, 0, stream>>>(h_B, cnt_B, tot4B);
}